// GCN_1056561954824
// MI455X (gfx1250) — compile-verified
//
#include <hip/hip_runtime.h>
#include <cstdint>
#include <cstddef>

typedef __attribute__((ext_vector_type(2))) float v2f;
typedef __attribute__((ext_vector_type(8))) float v8f;

#define DDIM 128
#define LP   132   // LDS row pitch: 128 + 4 pad -> half-wave 64-bit frag reads hit disjoint banks

// out[N,128] = act(X[N,128]) @ W[128,128]; act = tanh when APPLY_TANH (fuses layer-1 activation)
template <bool APPLY_TANH>
__global__ __launch_bounds__(256) void gemm_wmma(const float* __restrict__ X,
                                                 const float* __restrict__ W,
                                                 float* __restrict__ out,
                                                 int nrows) {
  __shared__ float Xs[128 * LP];   // X tile, row-major [row][k]
  __shared__ float Wt[128 * LP];   // W transposed,   [n][k]

  const int tid  = threadIdx.x;
  const int lane = tid & 31;
  const int wave = tid >> 5;      // 0..7
  const int half = lane >> 4;     // 0/1
  const int ln16 = lane & 15;
  const int rowBase = blockIdx.x * 128;

  // ---- stage X tile (coalesced float4 loads; OOB rows zero-filled) ----
  for (int i = 0; i < 16; ++i) {
    int idx = tid + i * 256;      // 4096 float4 slots
    int r   = idx >> 5;           // 0..127
    int c4  = idx & 31;           // 0..31
    int grow = rowBase + r;
    float4 v = make_float4(0.f, 0.f, 0.f, 0.f);
    if (grow < nrows) {
      v = *(const float4*)(X + (size_t)grow * DDIM + c4 * 4);
      if (APPLY_TANH) {
        v.x = tanhf(v.x); v.y = tanhf(v.y); v.z = tanhf(v.z); v.w = tanhf(v.w);
      }
    }
    float* p = &Xs[r * LP + c4 * 4];
    p[0] = v.x; p[1] = v.y; p[2] = v.z; p[3] = v.w;
  }
  // ---- stage W transposed (coalesced global read, scalar LDS transpose writes) ----
  for (int i = 0; i < 16; ++i) {
    int idx = tid + i * 256;
    int k   = idx >> 5;
    int c4  = idx & 31;
    float4 v = *(const float4*)(W + k * DDIM + c4 * 4);
    Wt[(c4 * 4 + 0) * LP + k] = v.x;
    Wt[(c4 * 4 + 1) * LP + k] = v.y;
    Wt[(c4 * 4 + 2) * LP + k] = v.z;
    Wt[(c4 * 4 + 3) * LP + k] = v.w;
  }
  __syncthreads();

  // wave 'w' computes rows [w*16, w*16+16) x 128 cols as 8 16x16 WMMA tiles
  const int mRow = wave * 16;
  v8f acc[8] = {};

  for (int k0 = 0; k0 < 128; k0 += 4) {
    // A frag 16x4: lane(ln16,half) holds X[mRow+ln16][k0 + 2*half + {0,1}]
    const float* ap = &Xs[(mRow + ln16) * LP + k0 + 2 * half];
    v2f a; a.x = ap[0]; a.y = ap[1];
#pragma unroll
    for (int t = 0; t < 8; ++t) {
      // B frag 4x16: lane(ln16,half) holds W[k0 + 2*half + {0,1}][t*16 + ln16]
      const float* bp = &Wt[(t * 16 + ln16) * LP + k0 + 2 * half];
      v2f b; b.x = bp[0]; b.y = bp[1];
      acc[t] = __builtin_amdgcn_wmma_f32_16x16x4_f32(false, a, false, b,
                                                     (short)0, acc[t], false, false);
    }
  }

  // C layout: VGPR r -> row (r + 8*half), col ln16
#pragma unroll
  for (int t = 0; t < 8; ++t) {
#pragma unroll
    for (int r = 0; r < 8; ++r) {
      int grow = rowBase + mRow + r + 8 * half;
      if (grow < nrows)
        out[(size_t)grow * DDIM + t * 16 + ln16] = acc[t][r];
    }
  }
}

// one wave32 per edge: gather h[src] (float4/lane), scale by ew, atomic scatter-add to out[dst]
__global__ __launch_bounds__(256) void edge_scatter(const float* __restrict__ h,
                                                    const int* __restrict__ src,
                                                    const int* __restrict__ dst,
                                                    const float* __restrict__ ew,
                                                    float* __restrict__ out,
                                                    int E) {
  int gwave = (int)((blockIdx.x * 256u + threadIdx.x) >> 5);
  int lane  = threadIdx.x & 31;
  if (gwave >= E) return;
  int   s = src[gwave];
  int   d = dst[gwave];
  float w = ew[gwave];
  float4 v = *(const float4*)(h + (size_t)s * DDIM + lane * 4);
  float* op = out + (size_t)d * DDIM + lane * 4;
  atomicAdd(op + 0, v.x * w);
  atomicAdd(op + 1, v.y * w);
  atomicAdd(op + 2, v.z * w);
  atomicAdd(op + 3, v.w * w);
}

__global__ __launch_bounds__(256) void tanh_inplace4(float* __restrict__ p, int n4) {
  int i = blockIdx.x * 256 + threadIdx.x;
  if (i < n4) {
    float4 v = ((float4*)p)[i];
    v.x = tanhf(v.x); v.y = tanhf(v.y); v.z = tanhf(v.z); v.w = tanhf(v.w);
    ((float4*)p)[i] = v;
  }
}

extern "C" void kernel_launch(void* const* d_in, const int* in_sizes, int n_in,
                              void* d_out, int out_size, void* d_ws, size_t ws_size,
                              hipStream_t stream) {
  const float* x   = (const float*)d_in[0];
  const int*   src = (const int*)  d_in[1];
  const int*   dst = (const int*)  d_in[2];
  const float* ew  = (const float*)d_in[3];
  const float* W1  = (const float*)d_in[4];
  const float* W2  = (const float*)d_in[5];
  float* out = (float*)d_out;

  const int N = in_sizes[0] / DDIM;
  const int E = in_sizes[1];

  float* h    = (float*)d_ws;              // N*128 f32
  float* acc1 = h + (size_t)N * DDIM;      // N*128 f32

  const int gblocks = (N + 127) / 128;
  const int eblocks = (E + 7) / 8;         // 8 waves / block, 1 edge / wave
  const int n4      = N * DDIM / 4;
  const size_t fbytes = (size_t)N * DDIM * sizeof(float);

  // layer 1: h = x @ W1 ; acc1 = segment_sum(h[src]*ew -> dst)
  gemm_wmma<false><<<gblocks, 256, 0, stream>>>(x, W1, h, N);
  hipMemsetAsync(acc1, 0, fbytes, stream);
  edge_scatter<<<eblocks, 256, 0, stream>>>(h, src, dst, ew, acc1, E);

  // layer 2: h = tanh(acc1) @ W2 ; out = tanh(segment_sum(h[src]*ew -> dst))
  gemm_wmma<true><<<gblocks, 256, 0, stream>>>(acc1, W2, h, N);
  hipMemsetAsync(out, 0, fbytes, stream);
  edge_scatter<<<eblocks, 256, 0, stream>>>(h, src, dst, ew, out, E);
  tanh_inplace4<<<(n4 + 255) / 256, 256, 0, stream>>>(out, n4);
}